// DnlsLoss_47588237639958
// MI455X (gfx1250) — compile-verified
//
#include <hip/hip_runtime.h>

typedef float v2f __attribute__((ext_vector_type(2)));
typedef float v8f __attribute__((ext_vector_type(8)));

#define B_    2
#define T_    5
#define C_    3
#define H_    96
#define W_    96
#define WS_   9
#define WT_   1
#define PS_   5
#define PSD_  7
#define K_    10
#define S0_   4
#define NH_   (H_ / S0_)            // 24
#define NW_   (W_ / S0_)            // 24
#define Q_    (T_ * NH_ * NW_)      // 2880
#define L_    ((2*WT_+1) * WS_ * WS_) // 243 candidates
#define LPAD_ 256
#define F1_   (C_ * PS_ * PS_)      // 75  (search feature dim)
#define F1P_  80                    // padded to multiple of 4 (WMMA K)
#define F2_   (C_ * PSD_ * PSD_)    // 147 (refine feature dim)
#define LSELF_ 121
#define DENOM_ (B_ * Q_ * (K_ - 1)) // 51840

__device__ __forceinline__ int reflect_(int i, int n) {
    i = i < 0 ? -i : i;
    return i >= n ? 2 * (n - 1) - i : i;
}
__device__ __forceinline__ int clip_(int i, int n) {
    return i < 0 ? 0 : (i > n - 1 ? n - 1 : i);
}
__device__ __forceinline__ int vidx_(int b, int t, int c, int y, int x) {
    return (((b * T_ + t) * C_ + c) * H_ + y) * W_ + x;
}

// One wave32 per (b, q): WMMA-based NLS distance + top-K + fp32 refinement.
__global__ __launch_bounds__(32)
void dnls_query_kernel(const float* __restrict__ noisy,
                       const float* __restrict__ deno,
                       float* __restrict__ partial)
{
    __shared__ float Aq[F1P_];        // query features (padded with zeros)
    __shared__ float Ac[16][F1P_];    // candidate tile features
    __shared__ float Dls[LPAD_];      // all candidate distances
    __shared__ float cnorm[16];
    __shared__ float qn_sh;
    __shared__ int   sel[K_];

    const int lane = threadIdx.x;
    const int bq = blockIdx.x;        // 0 .. B*Q-1
    const int b  = bq / Q_;
    const int q  = bq % Q_;
    const int qt = q / (NH_ * NW_);
    const int rem = q % (NH_ * NW_);
    const int qh = (rem / NW_) * S0_;
    const int qw = (rem % NW_) * S0_;

    // ---- load query patch features (reflect-padded 5x5x3) ----
    for (int f = lane; f < F1P_; f += 32) {
        float v = 0.0f;
        if (f < F1_) {
            int c = f / 25, ij = f % 25;
            int y = reflect_(qh + ij / 5 - 2, H_);
            int x = reflect_(qw + ij % 5 - 2, W_);
            v = noisy[vidx_(b, qt, c, y, x)];
        }
        Aq[f] = v;
    }
    for (int l = lane; l < LPAD_; l += 32) Dls[l] = __builtin_inff();
    __syncthreads();
    if (lane == 0) {
        float s = 0.0f;
        for (int f = 0; f < F1_; ++f) s += Aq[f] * Aq[f];
        qn_sh = s;
    }
    __syncthreads();
    const float qn = qn_sh;

    // ---- stage 1: 16 candidate tiles of 16; dot products via f32 WMMA ----
    const int m    = lane & 15;
    const int half = lane >> 4;            // 0: K{0,1}/M 0-7 ; 1: K{2,3}/M 8-15
    const int koff = half ? 2 : 0;

    for (int tile = 0; tile < 16; ++tile) {
        for (int e = lane; e < 16 * F1P_; e += 32) {
            int li = e / F1P_, f = e % F1P_;
            int l = tile * 16 + li;
            float v = 0.0f;
            if (l < L_ && f < F1_) {
                int dti = l / 81, r2 = l % 81;
                int ct = clip_(qt + dti - WT_, T_);
                int ch = clip_(qh + r2 / 9 - 4, H_);
                int cw = clip_(qw + r2 % 9 - 4, W_);
                int c = f / 25, ij = f % 25;
                int y = reflect_(ch + ij / 5 - 2, H_);
                int x = reflect_(cw + ij % 5 - 2, W_);
                v = noisy[vidx_(b, ct, c, y, x)];
            }
            Ac[li][f] = v;
        }
        __syncthreads();
        if (lane < 16) {
            float s = 0.0f;
            for (int f = 0; f < F1_; ++f) s += Ac[lane][f] * Ac[lane][f];
            cnorm[lane] = s;
        }
        __syncthreads();

        // D = A(16 cand x 80) * B(80 x 16 replicated query), fp32 accumulate
        v8f acc = {0.f, 0.f, 0.f, 0.f, 0.f, 0.f, 0.f, 0.f};
        #pragma unroll
        for (int kk = 0; kk < F1P_ / 4; ++kk) {
            int kb = kk * 4 + koff;
            v2f a  = { Ac[m][kb], Ac[m][kb + 1] };   // A: lane=M, vgprs=K
            v2f bb = { Aq[kb],    Aq[kb + 1]    };   // B: lane=N (query bcast)
            acc = __builtin_amdgcn_wmma_f32_16x16x4_f32(
                /*neg_a=*/false, a, /*neg_b=*/false, bb,
                /*c_mod=*/(short)0, acc, /*reuse_a=*/false, /*reuse_b=*/false);
        }
        // column N=0 (lanes 0 and 16) holds all 16 candidate dot products
        if (m == 0) {
            int mbase = half ? 8 : 0;
            #pragma unroll
            for (int r = 0; r < 8; ++r) {
                int l = tile * 16 + mbase + r;
                if (l < L_)
                    Dls[l] = qn + cnorm[mbase + r] - 2.0f * acc[r];
            }
        }
        __syncthreads();
    }
    if (lane == 0) Dls[LSELF_] = -1.0f;   // force self-match to front
    __syncthreads();

    // ---- top-K (ascending distance, lowest index wins ties) ----
    for (int k = 0; k < K_; ++k) {
        float bv = __builtin_inff(); int bi = LPAD_;
        #pragma unroll
        for (int j = 0; j < 8; ++j) {
            int e = lane * 8 + j;
            float v = Dls[e];
            if (v < bv || (v == bv && e < bi)) { bv = v; bi = e; }
        }
        for (int off = 16; off > 0; off >>= 1) {
            float ov = __shfl_xor(bv, off, 32);
            int   oi = __shfl_xor(bi, off, 32);
            if (ov < bv || (ov == bv && oi < bi)) { bv = ov; bi = oi; }
        }
        if (lane == 0) { sel[k] = bi; Dls[bi] = __builtin_inff(); }
        __syncthreads();
    }

    // ---- stage 2: fp32 refinement over 7x7x3 patches, k = 1..9 ----
    float s = 0.0f;
    for (int e = lane; e < (K_ - 1) * F2_; e += 32) {
        int k = e / F2_ + 1;
        int f = e % F2_;
        int l = sel[k];
        int dti = l / 81, r2 = l % 81;
        int ct = clip_(qt + dti - WT_, T_);
        int ch = clip_(qh + r2 / 9 - 4, H_);
        int cw = clip_(qw + r2 % 9 - 4, W_);
        int c = f / 49, ij = f % 49;
        int di = ij / 7 - 3, dj = ij % 7 - 3;
        float pd = deno [vidx_(b, qt, c, reflect_(qh + di, H_), reflect_(qw + dj, W_))];
        float pn = noisy[vidx_(b, ct, c, reflect_(ch + di, H_), reflect_(cw + dj, W_))];
        float dd = pd - pn;
        s += dd * dd;
    }
    for (int off = 16; off > 0; off >>= 1) s += __shfl_xor(s, off, 32);
    if (lane == 0) partial[blockIdx.x] = s;
}

// Deterministic final reduction (no float atomics -> replay-stable).
__global__ __launch_bounds__(256)
void dnls_reduce_kernel(const float* __restrict__ partial, int n,
                        float* __restrict__ out)
{
    __shared__ float sm[256];
    float s = 0.0f;
    for (int i = threadIdx.x; i < n; i += 256) s += partial[i];
    sm[threadIdx.x] = s;
    __syncthreads();
    for (int w = 128; w > 0; w >>= 1) {
        if (threadIdx.x < w) sm[threadIdx.x] += sm[threadIdx.x + w];
        __syncthreads();
    }
    if (threadIdx.x == 0) out[0] = sm[0] * (1.0f / (float)DENOM_);
}

extern "C" void kernel_launch(void* const* d_in, const int* in_sizes, int n_in,
                              void* d_out, int out_size, void* d_ws, size_t ws_size,
                              hipStream_t stream)
{
    const float* noisy = (const float*)d_in[0];
    const float* deno  = (const float*)d_in[1];
    // d_in[2] = curr_epoch (unused by the reference loss)
    float* partial = (float*)d_ws;          // B*Q floats = 23 KB scratch

    dnls_query_kernel<<<dim3(B_ * Q_), dim3(32), 0, stream>>>(noisy, deno, partial);
    dnls_reduce_kernel<<<dim3(1), dim3(256), 0, stream>>>(partial, B_ * Q_,
                                                          (float*)d_out);
}